// UnsupervisedGCN_309237645657
// MI455X (gfx1250) — compile-verified
//
#include <hip/hip_runtime.h>
#include <math.h>

#define N_NODES 100000
#define N_EDGES 1600000
#define HID 64
#define N_LAYERS 2
#define N_GRAPHS 256
#define S2S_ITERS 3
#define EPS 1e-5f

typedef float v2f __attribute__((ext_vector_type(2)));
typedef float v8f __attribute__((ext_vector_type(8)));

// ---------------------------------------------------------------- utilities
__global__ void fill_kernel(float* __restrict__ p, float v, int n) {
    int i = blockIdx.x * blockDim.x + threadIdx.x;
    if (i < n) p[i] = v;
}

__device__ __forceinline__ void atomicMaxFloat(float* addr, float v) {
    // standard bit-punning trick: valid for all float orderings, init to -inf
    if (v >= 0.0f) atomicMax((int*)addr, __float_as_int(v));
    else           atomicMin((unsigned int*)addr, __float_as_uint(v));
}

__device__ __forceinline__ float sigmoidf_(float x) {
    return 1.0f / (1.0f + expf(-x));
}

// ------------------------------------------------- edge score + segment max
// one wave (32 lanes) per edge; lane loads float2 -> 64-float dot
__global__ void edge_score_kernel(const float* __restrict__ h,
                                  const int* __restrict__ src,
                                  const int* __restrict__ dst,
                                  float* __restrict__ score,
                                  float* __restrict__ smax) {
    int e    = (blockIdx.x * blockDim.x + threadIdx.x) >> 5;
    int lane = threadIdx.x & 31;
    if (e >= N_EDGES) return;
    int s = src[e], d = dst[e];
    float2 a = ((const float2*)(h + (size_t)s * HID))[lane];
    float2 b = ((const float2*)(h + (size_t)d * HID))[lane];
    float v = a.x * b.x + a.y * b.y;
    for (int off = 16; off > 0; off >>= 1) v += __shfl_down(v, off, 32);
    if (lane == 0) {
        score[e] = v;
        atomicMaxFloat(&smax[d], v);
    }
}

// ------------------------------------------------- edge softmax numerator + denom
__global__ void edge_exp_kernel(const int* __restrict__ dst,
                                float* __restrict__ score,
                                const float* __restrict__ smax,
                                float* __restrict__ denom) {
    int e = blockIdx.x * blockDim.x + threadIdx.x;
    if (e >= N_EDGES) return;
    int d = dst[e];
    float ex = expf(score[e] - smax[d]);
    score[e] = ex;
    atomicAdd(&denom[d], ex);
}

// ------------------------------------------------- dual GEMM via f32 WMMA
// ft = h @ Wc ; res = relu(h @ Wr + br). One wave -> 16 rows x 64 cols,
// accumulating K=64 with 16x16x4 f32 WMMA. A fragment shared across both
// weight matrices and all 4 N-tiles.
__global__ __launch_bounds__(128) void gemm_dual_kernel(
        const float* __restrict__ A,
        const float* __restrict__ Wc,
        const float* __restrict__ Wr,
        const float* __restrict__ br,
        float* __restrict__ ft,
        float* __restrict__ res) {
    const int wave = threadIdx.x >> 5;
    const int lane = threadIdx.x & 31;
    const int mt   = blockIdx.x * 4 + wave;           // 16-row tile index
    if (mt * 16 >= N_NODES) return;                   // wave-uniform guard
    const int mbase = mt * 16;
    const int half  = lane >> 4;                      // 0 | 1
    const int l16   = lane & 15;
    const int kb    = half * 2;

    v8f accC[4] = {(v8f)0.f, (v8f)0.f, (v8f)0.f, (v8f)0.f};
    v8f accR[4] = {(v8f)0.f, (v8f)0.f, (v8f)0.f, (v8f)0.f};

    for (int kk = 0; kk < HID; kk += 4) {
        // A 16x4 fragment: lane l16 holds row m, Ks kb..kb+1
        const float* arow = A + (size_t)(mbase + l16) * HID + kk + kb;
        v2f a; a.x = arow[0]; a.y = arow[1];
#pragma unroll
        for (int nt = 0; nt < 4; ++nt) {
            const int n = nt * 16 + l16;
            v2f bc, bg;
            bc.x = Wc[(kk + kb) * HID + n];
            bc.y = Wc[(kk + kb + 1) * HID + n];
            bg.x = Wr[(kk + kb) * HID + n];
            bg.y = Wr[(kk + kb + 1) * HID + n];
            accC[nt] = __builtin_amdgcn_wmma_f32_16x16x4_f32(
                false, a, false, bc, (short)0, accC[nt], false, false);
            accR[nt] = __builtin_amdgcn_wmma_f32_16x16x4_f32(
                false, a, false, bg, (short)0, accR[nt], false, false);
        }
    }
#pragma unroll
    for (int nt = 0; nt < 4; ++nt) {
        const int n = nt * 16 + l16;
        const float bias = br[n];
#pragma unroll
        for (int r = 0; r < 8; ++r) {
            const int row = mbase + r + half * 8;
            ft[(size_t)row * HID + n] = accC[nt][r];
            float v = accR[nt][r] + bias;
            res[(size_t)row * HID + n] = v > 0.0f ? v : 0.0f;
        }
    }
}

// ------------------------------------------------- attention-weighted scatter-add
__global__ void aggregate_kernel(const int* __restrict__ src,
                                 const int* __restrict__ dst,
                                 const float* __restrict__ ex,
                                 const float* __restrict__ denom,
                                 const float* __restrict__ ft,
                                 float* __restrict__ agg) {
    int e    = (blockIdx.x * blockDim.x + threadIdx.x) >> 5;
    int lane = threadIdx.x & 31;
    if (e >= N_EDGES) return;
    int s = src[e], d = dst[e];
    float coef = ex[e] / (denom[d] + 1e-9f);
    float2 f = ((const float2*)(ft + (size_t)s * HID))[lane];
    atomicAdd(&agg[(size_t)d * HID + 2 * lane],     f.x * coef);
    atomicAdd(&agg[(size_t)d * HID + 2 * lane + 1], f.y * coef);
}

// ------------------------------------------------- out = agg + b_conv + res; BN stats
__global__ __launch_bounds__(256) void combine_stats_kernel(
        const float* __restrict__ agg,
        const float* __restrict__ res,
        const float* __restrict__ bconv,
        float* __restrict__ outbuf,
        float* __restrict__ stats /* sum[64], sumsq[64] */) {
    __shared__ float ssum[256], ssq[256];
    const int t   = threadIdx.x;
    const int c   = t & 63;
    const int sub = t >> 6;   // 0..3
    const float bc = bconv[c];
    float lsum = 0.0f, lsq = 0.0f;
    for (int node = blockIdx.x * 4 + sub; node < N_NODES; node += gridDim.x * 4) {
        size_t idx = (size_t)node * HID + c;
        float v = agg[idx] + bc + res[idx];
        outbuf[idx] = v;
        lsum += v; lsq += v * v;
    }
    ssum[t] = lsum; ssq[t] = lsq;
    __syncthreads();
    if (sub == 0) {
        float s = ssum[c] + ssum[64 + c] + ssum[128 + c] + ssum[192 + c];
        float q = ssq[c]  + ssq[64 + c]  + ssq[128 + c]  + ssq[192 + c];
        atomicAdd(&stats[c], s);
        atomicAdd(&stats[64 + c], q);
    }
}

// stats -> scale/shift (in place)
__global__ void finalize_stats_kernel(float* __restrict__ stats,
                                      const float* __restrict__ gamma,
                                      const float* __restrict__ beta) {
    int c = threadIdx.x;
    if (c >= HID) return;
    float mean = stats[c] / (float)N_NODES;
    float var  = stats[64 + c] / (float)N_NODES - mean * mean;
    float scale = gamma[c] * rsqrtf(var + EPS);
    stats[c]      = scale;
    stats[64 + c] = beta[c] - mean * scale;
}

__global__ void normalize_kernel(const float* __restrict__ outbuf,
                                 const float* __restrict__ stats,
                                 float* __restrict__ h) {
    int i = blockIdx.x * blockDim.x + threadIdx.x;
    if (i >= N_NODES * HID) return;
    int c = i & 63;
    h[i] = outbuf[i] * stats[c] + stats[64 + c];
}

// ------------------------------------------------- per-graph sum readout
__global__ void readout_kernel(const float* __restrict__ h,
                               const int* __restrict__ graph_ids,
                               float* __restrict__ hg) {
    int node = (blockIdx.x * blockDim.x + threadIdx.x) >> 5;
    int lane = threadIdx.x & 31;
    if (node >= N_NODES) return;
    int g = graph_ids[node];
    float2 v = ((const float2*)(h + (size_t)node * HID))[lane];
    atomicAdd(&hg[(size_t)g * HID + 2 * lane],     v.x);
    atomicAdd(&hg[(size_t)g * HID + 2 * lane + 1], v.y);
}

// ------------------------------------------------- Set2Set (LSTM+attn) + LayerNorm
__global__ __launch_bounds__(256) void set2set_kernel(
        const float* __restrict__ hg,
        const float* __restrict__ W_ih,   // [256,128]
        const float* __restrict__ W_hh,   // [256,64]
        const float* __restrict__ b_lstm, // [256]
        const float* __restrict__ ln_gamma,
        const float* __restrict__ ln_beta,
        float* __restrict__ out) {
    __shared__ float q_star[2 * HID], hl[HID], cl[HID];
    __shared__ float gates[256], logit[256], red[256];
    const int t = threadIdx.x;
    if (t < 2 * HID) q_star[t] = 0.0f;
    if (t < HID) { hl[t] = 0.0f; cl[t] = 0.0f; }
    __syncthreads();

    for (int it = 0; it < S2S_ITERS; ++it) {
        // gates = W_ih @ q_star + W_hh @ hl + b
        float g = b_lstm[t];
        for (int k = 0; k < 2 * HID; ++k) g += W_ih[t * 2 * HID + k] * q_star[k];
        for (int k = 0; k < HID; ++k)     g += W_hh[t * HID + k] * hl[k];
        gates[t] = g;
        __syncthreads();
        if (t < HID) {
            float i_ = sigmoidf_(gates[t]);
            float f_ = sigmoidf_(gates[HID + t]);
            float g_ = tanhf(gates[2 * HID + t]);
            float o_ = sigmoidf_(gates[3 * HID + t]);
            float c = f_ * cl[t] + i_ * g_;
            cl[t] = c;
            hl[t] = o_ * tanhf(c);
        }
        __syncthreads();
        // logits over 256 graphs
        float lv = 0.0f;
        for (int k = 0; k < HID; ++k) lv += hg[t * HID + k] * hl[k];
        logit[t] = lv; red[t] = lv;
        __syncthreads();
        for (int s = 128; s > 0; s >>= 1) {
            if (t < s) red[t] = fmaxf(red[t], red[t + s]);
            __syncthreads();
        }
        float mx = red[0];
        __syncthreads();
        float ex = expf(logit[t] - mx);
        logit[t] = ex; red[t] = ex;
        __syncthreads();
        for (int s = 128; s > 0; s >>= 1) {
            if (t < s) red[t] += red[t + s];
            __syncthreads();
        }
        float dsum = red[0];
        __syncthreads();
        if (t < HID) {
            float acc = 0.0f;
            for (int g2 = 0; g2 < N_GRAPHS; ++g2) acc += logit[g2] * hg[g2 * HID + t];
            float r = acc / dsum;
            q_star[t] = hl[t];
            q_star[HID + t] = r;
        }
        __syncthreads();
    }
    // LayerNorm over q_star[128]
    red[t] = (t < 2 * HID) ? q_star[t] : 0.0f;
    __syncthreads();
    for (int s = 128; s > 0; s >>= 1) { if (t < s) red[t] += red[t + s]; __syncthreads(); }
    float mu = red[0] / (float)(2 * HID);
    __syncthreads();
    float dvv = (t < 2 * HID) ? (q_star[t] - mu) : 0.0f;
    red[t] = dvv * dvv;
    __syncthreads();
    for (int s = 128; s > 0; s >>= 1) { if (t < s) red[t] += red[t + s]; __syncthreads(); }
    float var = red[0] / (float)(2 * HID);
    if (t < 2 * HID)
        out[t] = (q_star[t] - mu) * rsqrtf(var + EPS) * ln_gamma[t] + ln_beta[t];
}

// ---------------------------------------------------------------- launcher
extern "C" void kernel_launch(void* const* d_in, const int* in_sizes, int n_in,
                              void* d_out, int out_size, void* d_ws, size_t ws_size,
                              hipStream_t stream) {
    const float* x        = (const float*)d_in[0];
    const float* W_conv   = (const float*)d_in[1];
    const float* b_conv   = (const float*)d_in[2];
    const float* W_res    = (const float*)d_in[3];
    const float* b_res    = (const float*)d_in[4];
    const float* bn_gamma = (const float*)d_in[5];
    const float* bn_beta  = (const float*)d_in[6];
    const float* W_ih     = (const float*)d_in[7];
    const float* W_hh     = (const float*)d_in[8];
    const float* b_lstm   = (const float*)d_in[9];
    const float* ln_gamma = (const float*)d_in[10];
    const float* ln_beta  = (const float*)d_in[11];
    const int*   src      = (const int*)d_in[12];
    const int*   dst      = (const int*)d_in[13];
    const int*   gids     = (const int*)d_in[14];

    const size_t NH = (size_t)N_NODES * HID;
    float* ws    = (float*)d_ws;
    float* h     = ws;             // NH
    float* ft    = h + NH;         // NH  (reused as BN input "out" buffer)
    float* res   = ft + NH;        // NH
    float* agg   = res + NH;       // NH
    float* score = agg + NH;       // E
    float* smax  = score + N_EDGES;  // N
    float* denom = smax + N_NODES;   // N
    float* stats = denom + N_NODES;  // 128
    float* hg    = stats + 128;      // 256*64

    const int edgeWarpBlocks = (N_EDGES * 32 + 255) / 256;
    const int edgeThrBlocks  = (N_EDGES + 255) / 256;
    const int nodeWarpBlocks = (N_NODES * 32 + 255) / 256;
    const int gemmBlocks     = (N_NODES / 16 + 3) / 4;   // 6250 tiles / 4 waves

    for (int l = 0; l < N_LAYERS; ++l) {
        const float* hin = (l == 0) ? x : h;
        const float* Wc = W_conv + (size_t)l * HID * HID;
        const float* Wr = W_res  + (size_t)l * HID * HID;

        fill_kernel<<<(N_NODES + 255) / 256, 256, 0, stream>>>(smax, -__builtin_inff(), N_NODES);
        fill_kernel<<<(N_NODES + 255) / 256, 256, 0, stream>>>(denom, 0.0f, N_NODES);
        fill_kernel<<<((int)NH + 255) / 256, 256, 0, stream>>>(agg, 0.0f, (int)NH);
        fill_kernel<<<1, 256, 0, stream>>>(stats, 0.0f, 128);

        edge_score_kernel<<<edgeWarpBlocks, 256, 0, stream>>>(hin, src, dst, score, smax);
        edge_exp_kernel<<<edgeThrBlocks, 256, 0, stream>>>(dst, score, smax, denom);
        gemm_dual_kernel<<<gemmBlocks, 128, 0, stream>>>(hin, Wc, Wr, b_res + l * HID, ft, res);
        aggregate_kernel<<<edgeWarpBlocks, 256, 0, stream>>>(src, dst, score, denom, ft, agg);
        combine_stats_kernel<<<512, 256, 0, stream>>>(agg, res, b_conv + l * HID, ft, stats);
        finalize_stats_kernel<<<1, 64, 0, stream>>>(stats, bn_gamma + l * HID, bn_beta + l * HID);
        normalize_kernel<<<((int)NH + 255) / 256, 256, 0, stream>>>(ft, stats, h);
    }

    fill_kernel<<<(N_GRAPHS * HID + 255) / 256, 256, 0, stream>>>(hg, 0.0f, N_GRAPHS * HID);
    readout_kernel<<<nodeWarpBlocks, 256, 0, stream>>>(h, gids, hg);
    set2set_kernel<<<1, 256, 0, stream>>>(hg, W_ih, W_hh, b_lstm, ln_gamma, ln_beta, (float*)d_out);
}